// LIF_1932735283690
// MI455X (gfx1250) — compile-verified
//
#include <hip/hip_runtime.h>

// Multistep LIF forward, hard reset, tau=0.5, vthr=1.0.
// x      : (T=4, N) f32, N = 32*512*1024
// d_out  : [ spikes(T*N) | mems(T*N) ] f32
//
// HBM-streaming kernel: 48 B of traffic per spatial lane -> ~805 MB total,
// ~35 us floor at 23.3 TB/s. All 4 timestep loads kept in flight per thread
// (sched_barrier pins the schedule), b128 NT loads/stores, scan in registers.
// Uniform per-timestep base pointers keep address math on the SALU and let
// every access use the saddr + voffset(scale_offset) form. Wave32 / gfx1250.

typedef float v4f __attribute__((ext_vector_type(4)));

#define LIF_T    4
#define LIF_VTHR 1.0f
#define LIF_TAU  0.5f

__global__ __launch_bounds__(256) void lif_multistep_fwd(
    const v4f* __restrict__ x,
    v4f* __restrict__ spikes,
    v4f* __restrict__ mems,
    unsigned n4)  // N / 4 : number of float4 packets per timestep
{
    unsigned i = blockIdx.x * 256u + threadIdx.x;
    if (i >= n4) return;

    // ---- Load phase: 4 outstanding global_load_b128 (NT) before any use ----
    v4f xv[LIF_T];
    {
        const v4f* __restrict__ xp = x;   // wave-uniform base, stepped on SALU
#pragma unroll
        for (int t = 0; t < LIF_T; ++t) {
            xv[t] = __builtin_nontemporal_load(xp + i);
            xp += n4;
        }
    }
    // Keep all loads issued before compute is scheduled (MLP = 4).
    __builtin_amdgcn_sched_barrier(0);

    // ---- Scan phase: registers only ----
    v4f* __restrict__ sp = spikes;        // uniform bases, stepped on SALU
    v4f* __restrict__ mp = mems;
    v4f vm = {0.0f, 0.0f, 0.0f, 0.0f};
#pragma unroll
    for (int t = 0; t < LIF_T; ++t) {
        v4f v = vm + xv[t];               // integrate
        v4f s, vp;
#pragma unroll
        for (int c = 0; c < 4; ++c) {
            // spike = Heaviside(v - vthr); hard reset: v *= (1 - spike)
            bool fire = v[c] > LIF_VTHR;  // v_cmp_gt_f32
            s[c]  = fire ? 1.0f : 0.0f;   // v_cndmask_b32
            vp[c] = fire ? 0.0f : v[c];   // v_cndmask_b32
        }
        // Streaming (NT) stores: write-once outputs, > 192 MB L2 capacity.
        __builtin_nontemporal_store(s,  sp + i);
        __builtin_nontemporal_store(vp, mp + i);
        sp += n4;
        mp += n4;
        vm = vp * LIF_TAU;                // decay carried state
    }
}

extern "C" void kernel_launch(void* const* d_in, const int* in_sizes, int n_in,
                              void* d_out, int out_size, void* d_ws, size_t ws_size,
                              hipStream_t stream)
{
    const float* x   = (const float*)d_in[0];
    float*       out = (float*)d_out;

    const long long totalTN = in_sizes[0];     // T * N = 67,108,864
    const long long N       = totalTN / LIF_T; // 16,777,216 (divisible by 4)
    const unsigned  n4      = (unsigned)(N / 4);

    float* spikes = out;             // first T*N floats
    float* mems   = out + totalTN;   // next  T*N floats

    const unsigned block = 256;
    const unsigned grid  = (n4 + block - 1) / block;  // 16384 blocks

    lif_multistep_fwd<<<dim3(grid), dim3(block), 0, stream>>>(
        (const v4f*)x, (v4f*)spikes, (v4f*)mems, n4);
}